// GBottleneck_71305047048354
// MI455X (gfx1250) — compile-verified
//
#include <hip/hip_runtime.h>
#include <hip/hip_bf16.h>

typedef float v2f __attribute__((ext_vector_type(2)));
typedef float v8f __attribute__((ext_vector_type(8)));

#define N_NODES 50000
#define N_EDGES 800000
#define HID 192
#define IN_DIM 3
#define OUT_DIM 3
#define N_BLOCKS 6
#define M_TILES (N_NODES / 16)   /* 3125 */
#define N_TILES (HID / 16)       /* 12   */
#define NH (N_NODES * HID)       /* 9,600,000 */

#define CHUNK_K   32                  /* K-rows staged per LDS chunk            */
#define N_CHUNKS  (HID / CHUNK_K)     /* 6                                      */
#define KK_PER_CHUNK (CHUNK_K / 4)    /* 8 kk-steps per chunk                   */
#define LDS_STRIDE 200                /* 192 + 8: rows r and r+2 hit disjoint   */
                                      /* 16-bank groups (2*200 mod 64 == 16)    */

// ---------------------------------------------------------------------------
// Dense GEMM: out[M,192] = in[M,192] @ w[192,192] (+ bias broadcast per col)
// One wave32 per 16-row M tile, 8 tiles per block. Weight matrix staged
// through LDS in 32-row chunks shared by all 8 waves of the workgroup.
// V_WMMA_F32_16X16X4_F32: fp32 in / fp32 accumulate (precision-exact).
// ---------------------------------------------------------------------------
__global__ __launch_bounds__(256) void wmma_gemm_192(
    const float* __restrict__ in, const float* __restrict__ w,
    const float* __restrict__ bias, float* __restrict__ out) {
  __shared__ float wlds[CHUNK_K * LDS_STRIDE];   // 25.6 KB

  const int tid   = threadIdx.x;
  const int wvblk = tid >> 5;
  const int lane  = tid & 31;
  const int l16   = lane & 15;
  const int half  = lane >> 4;

  int tile = blockIdx.x * 8 + wvblk;
  const bool live = (tile < M_TILES);
  if (!live) tile = M_TILES - 1;        // clamp: keep wave alive for barriers
  const int row = tile * 16 + l16;      // A-matrix M index for this lane

  v8f acc[N_TILES];
#pragma unroll
  for (int n = 0; n < N_TILES; ++n) {
    float bv = bias ? bias[n * 16 + l16] : 0.0f;  // C(M,N) init depends on N only
    acc[n] = (v8f){bv, bv, bv, bv, bv, bv, bv, bv};
  }

  const float* arow = in + (long long)row * HID;

  for (int ch = 0; ch < N_CHUNKS; ++ch) {
    __syncthreads();
    // ---- cooperative stage of 32x192 weight chunk into padded LDS ----
    // 1536 float4 groups; 256 threads x 6 groups each.
#pragma unroll
    for (int i = 0; i < 6; ++i) {
      int g = i * 256 + tid;            // group id
      int r = g / 48;                   // row within chunk (48 float4 per row)
      int c = (g - r * 48) * 4;         // starting col
      float4 v = *(const float4*)(w + (long long)(ch * CHUNK_K + r) * HID + c);
      *(float4*)(&wlds[r * LDS_STRIDE + c]) = v;
    }
    __syncthreads();

    // ---- 8 kk-steps consuming the chunk ----
    for (int kl = 0; kl < KK_PER_CHUNK; ++kl) {
      // A 16x4 f32 layout: VGPR0/1 = K {0,1} lanes 0-15, K {2,3} lanes 16-31
      int klds  = kl * 4 + 2 * half;            // row within LDS chunk
      int kbase = ch * CHUNK_K + klds;          // absolute K
      v2f a;
      a.x = arow[kbase];
      a.y = arow[kbase + 1];
#pragma unroll
      for (int n = 0; n < N_TILES; ++n) {
        int col = n * 16 + l16;
        v2f b;                                   // B mirrors A layout
        b.x = wlds[klds * LDS_STRIDE + col];
        b.y = wlds[(klds + 1) * LDS_STRIDE + col];
        acc[n] = __builtin_amdgcn_wmma_f32_16x16x4_f32(
            false, a, false, b, (short)0, acc[n], false, false);
      }
    }
  }

  if (!live) return;                    // wave-uniform; no WMMA past this point
  // C/D layout: VGPR v -> M = v + 8*half, N = l16
#pragma unroll
  for (int n = 0; n < N_TILES; ++n) {
    int col = n * 16 + l16;
#pragma unroll
    for (int v = 0; v < 8; ++v) {
      int r = tile * 16 + v + 8 * half;
      out[(long long)r * HID + col] = acc[n][v];
    }
  }
}

// ---------------------------------------------------------------------------
// Input head: sup = x @ W_in ; lin = x @ Wl_in + b_in  (K = 3, trivial FLOPs)
// ---------------------------------------------------------------------------
__global__ void gconv_in_kernel(const float* __restrict__ x,
                                const float* __restrict__ W,
                                const float* __restrict__ Wl,
                                const float* __restrict__ b,
                                float* __restrict__ sup,
                                float* __restrict__ lin) {
  int gid = blockIdx.x * blockDim.x + threadIdx.x;
  if (gid >= NH) return;
  int n = gid / HID, c = gid - n * HID;
  float x0 = x[n * 3 + 0], x1 = x[n * 3 + 1], x2 = x[n * 3 + 2];
  sup[gid] = x0 * W[c] + x1 * W[HID + c] + x2 * W[2 * HID + c];
  lin[gid] = x0 * Wl[c] + x1 * Wl[HID + c] + x2 * Wl[2 * HID + c] + b[c];
}

// ---------------------------------------------------------------------------
// Output head: sup3 = h @ W_out ; x_out = h @ Wl_out + b_out (N = 3)
// ---------------------------------------------------------------------------
__global__ void gconv_out_kernel(const float* __restrict__ h,
                                 const float* __restrict__ W,
                                 const float* __restrict__ Wl,
                                 const float* __restrict__ b,
                                 float* __restrict__ sup,
                                 float* __restrict__ xout) {
  int gid = blockIdx.x * blockDim.x + threadIdx.x;
  if (gid >= N_NODES * OUT_DIM) return;
  int n = gid / OUT_DIM, c = gid - n * OUT_DIM;
  const float* hr = h + (long long)n * HID;
  float s = 0.f, l = 0.f;
  for (int k = 0; k < HID; ++k) {
    float hv = hr[k];
    s += hv * W[k * OUT_DIM + c];
    l += hv * Wl[k * OUT_DIM + c];
  }
  sup[gid]  = s;
  xout[gid] = l + b[c];
}

// ---------------------------------------------------------------------------
// Edge scatter (192 ch): out[dst] += ew * sup[src], float4 per thread.
// ---------------------------------------------------------------------------
__global__ void scatter192_kernel(const float* __restrict__ sup,
                                  float* __restrict__ out,
                                  const int* __restrict__ src,
                                  const int* __restrict__ dst,
                                  const float* __restrict__ ew) {
  int gid = blockIdx.x * blockDim.x + threadIdx.x;
  if (gid >= N_EDGES * 48) return;
  int e  = gid / 48;
  int c4 = (gid - e * 48) * 4;
  float w = ew[e];
  const float4 v = *(const float4*)(sup + (long long)src[e] * HID + c4);
  float* o = out + (long long)dst[e] * HID + c4;
  __hip_atomic_fetch_add(o + 0, w * v.x, __ATOMIC_RELAXED, __HIP_MEMORY_SCOPE_AGENT);
  __hip_atomic_fetch_add(o + 1, w * v.y, __ATOMIC_RELAXED, __HIP_MEMORY_SCOPE_AGENT);
  __hip_atomic_fetch_add(o + 2, w * v.z, __ATOMIC_RELAXED, __HIP_MEMORY_SCOPE_AGENT);
  __hip_atomic_fetch_add(o + 3, w * v.w, __ATOMIC_RELAXED, __HIP_MEMORY_SCOPE_AGENT);
}

// Edge scatter, 3 output channels.
__global__ void scatter3_kernel(const float* __restrict__ sup,
                                float* __restrict__ out,
                                const int* __restrict__ src,
                                const int* __restrict__ dst,
                                const float* __restrict__ ew) {
  int gid = blockIdx.x * blockDim.x + threadIdx.x;
  if (gid >= N_EDGES * OUT_DIM) return;
  int e = gid / OUT_DIM, c = gid - e * OUT_DIM;
  float v = ew[e] * sup[(long long)src[e] * OUT_DIM + c];
  __hip_atomic_fetch_add(out + (long long)dst[e] * OUT_DIM + c, v,
                         __ATOMIC_RELAXED, __HIP_MEMORY_SCOPE_AGENT);
}

__global__ void relu_kernel(float* __restrict__ p, int n) {
  int i = blockIdx.x * blockDim.x + threadIdx.x;
  if (i < n) p[i] = fmaxf(p[i], 0.0f);
}

// h = (h + relu(y)) * 0.5   (fused tail of each GResBlock)
__global__ void residual_relu_kernel(float* __restrict__ h,
                                     const float* __restrict__ y, int n) {
  int i = blockIdx.x * blockDim.x + threadIdx.x;
  if (i < n) h[i] = (h[i] + fmaxf(y[i], 0.0f)) * 0.5f;
}

// ---------------------------------------------------------------------------
extern "C" void kernel_launch(void* const* d_in, const int* in_sizes, int n_in,
                              void* d_out, int out_size, void* d_ws, size_t ws_size,
                              hipStream_t stream) {
  const float* x      = (const float*)d_in[0];
  const int*   src    = (const int*)  d_in[1];
  const int*   dst    = (const int*)  d_in[2];
  const float* ew     = (const float*)d_in[3];
  const float* W_in   = (const float*)d_in[4];
  const float* Wl_in  = (const float*)d_in[5];
  const float* b_in   = (const float*)d_in[6];
  const float* W_blk  = (const float*)d_in[7];   // [6,2,192,192]
  const float* Wl_blk = (const float*)d_in[8];
  const float* b_blk  = (const float*)d_in[9];   // [6,2,192]
  const float* W_out  = (const float*)d_in[10];
  const float* Wl_out = (const float*)d_in[11];
  const float* b_out  = (const float*)d_in[12];

  float* xout = (float*)d_out;                   // [50000,3]
  float* h    = xout + N_NODES * OUT_DIM;        // x_hidden lives in d_out

  float* S  = (float*)d_ws;                      // sup scratch  [50000,192]
  float* Y1 = S  + NH;
  float* Y2 = Y1 + NH;

  const int EW_BLK = 256;
  const int g_nh   = (NH + 255) / 256;                 // 37500
  const int g_gemm = (M_TILES + 7) / 8;                // 391 blocks of 8 waves
  const int g_sc   = (N_EDGES * 48 + 255) / 256;       // 150000
  const int g_sc3  = (N_EDGES * OUT_DIM + 255) / 256;

  // ---- conv1 + ReLU ----
  gconv_in_kernel<<<g_nh, EW_BLK, 0, stream>>>(x, W_in, Wl_in, b_in, S, h);
  scatter192_kernel<<<g_sc, EW_BLK, 0, stream>>>(S, h, src, dst, ew);
  relu_kernel<<<g_nh, EW_BLK, 0, stream>>>(h, NH);

  // ---- 6 GResBlocks ----
  for (int blk = 0; blk < N_BLOCKS; ++blk) {
    const float* W1  = W_blk  + (size_t)(blk * 2 + 0) * HID * HID;
    const float* W2  = W_blk  + (size_t)(blk * 2 + 1) * HID * HID;
    const float* Wl1 = Wl_blk + (size_t)(blk * 2 + 0) * HID * HID;
    const float* Wl2 = Wl_blk + (size_t)(blk * 2 + 1) * HID * HID;
    const float* b1  = b_blk  + (size_t)(blk * 2 + 0) * HID;
    const float* b2  = b_blk  + (size_t)(blk * 2 + 1) * HID;

    // y1 = relu(gconv(h, W1, Wl1, b1))
    wmma_gemm_192<<<g_gemm, 256, 0, stream>>>(h, W1, nullptr, S);
    wmma_gemm_192<<<g_gemm, 256, 0, stream>>>(h, Wl1, b1, Y1);
    scatter192_kernel<<<g_sc, EW_BLK, 0, stream>>>(S, Y1, src, dst, ew);
    relu_kernel<<<g_nh, EW_BLK, 0, stream>>>(Y1, NH);

    // y2 = gconv(y1, W2, Wl2, b2); h = (h + relu(y2)) * 0.5
    wmma_gemm_192<<<g_gemm, 256, 0, stream>>>(Y1, W2, nullptr, S);
    wmma_gemm_192<<<g_gemm, 256, 0, stream>>>(Y1, Wl2, b2, Y2);
    scatter192_kernel<<<g_sc, EW_BLK, 0, stream>>>(S, Y2, src, dst, ew);
    residual_relu_kernel<<<g_nh, EW_BLK, 0, stream>>>(h, Y2, NH);
  }

  // ---- conv2 (output head, no activation). h already sits in d_out. ----
  gconv_out_kernel<<<(N_NODES * OUT_DIM + 255) / 256, EW_BLK, 0, stream>>>(
      h, W_out, Wl_out, b_out, S, xout);
  scatter3_kernel<<<g_sc3, EW_BLK, 0, stream>>>(S, xout, src, dst, ew);
}